// RGCNLowMem_3908420239948
// MI455X (gfx1250) — compile-verified
//
#include <hip/hip_runtime.h>

typedef __attribute__((ext_vector_type(2))) float v2f;
typedef __attribute__((ext_vector_type(8))) float v8f;

#define D 64
#define NREL 16

// Workspace layout:
//   int32 region:
//     [ 0..15]  rel_count[16]
//     [16..32]  rel_off[17]   (exclusive prefix of counts)
//     [33..49]  tile_off[17]  (exclusive prefix of ceil(count/16))
//     [50..65]  cursor[16]
//     [66.. ]   sorted_eid[E]
//   then (8B aligned): packed weight fragments, NREL*16*4*32 v2f (256 KB)
#define WS_CNT     0
#define WS_RELOFF  16
#define WS_TILEOFF 33
#define WS_CURSOR  50
#define WS_SORTED  66

__global__ void rgcn_zero_f32(float* __restrict__ p, int n) {
    int i = blockIdx.x * blockDim.x + threadIdx.x;
    int stride = gridDim.x * blockDim.x;
    for (; i < n; i += stride) p[i] = 0.0f;
}

__global__ void rgcn_zero_i32(int* __restrict__ p, int n) {
    int i = blockIdx.x * blockDim.x + threadIdx.x;
    if (i < n) p[i] = 0;
}

__global__ void rgcn_hist(const int* __restrict__ etypes, int E, int* __restrict__ ws) {
    int i = blockIdx.x * blockDim.x + threadIdx.x;
    if (i < E) {
        int r = etypes[i] & (NREL - 1);
        atomicAdd(&ws[WS_CNT + r], 1);
    }
}

__global__ void rgcn_scan(int* __restrict__ ws) {
    if (blockIdx.x == 0 && threadIdx.x == 0) {
        int off = 0, toff = 0;
        for (int r = 0; r < NREL; ++r) {
            ws[WS_RELOFF + r]  = off;
            ws[WS_TILEOFF + r] = toff;
            int c = ws[WS_CNT + r];
            off  += c;
            toff += (c + 15) >> 4;
            ws[WS_CURSOR + r] = 0;
        }
        ws[WS_RELOFF + NREL]  = off;
        ws[WS_TILEOFF + NREL] = toff;
    }
}

__global__ void rgcn_scatter(const int* __restrict__ etypes, int E, int* __restrict__ ws) {
    int i = blockIdx.x * blockDim.x + threadIdx.x;
    if (i < E) {
        int r = etypes[i] & (NREL - 1);
        int pos = atomicAdd(&ws[WS_CURSOR + r], 1);
        ws[WS_SORTED + ws[WS_RELOFF + r] + pos] = i;
    }
}

// Pre-pack weights into per-lane WMMA B fragments:
//   packed[((r*16 + k)*4 + n)*32 + lane] = { W[r][kk][col], W[r][kk+1][col] }
// with kk = k*4 + (lane>=16 ? 2 : 0), col = n*16 + (lane&15).
// Makes every B fragment a single coalesced global_load_b64.
__global__ void rgcn_pack_w(const float* __restrict__ w, v2f* __restrict__ packed) {
    int i = blockIdx.x * blockDim.x + threadIdx.x;   // NREL*16*4*32 = 32768
    if (i >= NREL * 16 * 4 * 32) return;
    int lane = i & 31;
    int n    = (i >> 5) & 3;
    int k    = (i >> 7) & 15;
    int r    =  i >> 11;
    int kk   = k * 4 + ((lane >= 16) ? 2 : 0);
    int col  = n * 16 + (lane & 15);
    const float* W = w + (long)r * D * D;
    v2f b;
    b.x = W[kk * D + col];
    b.y = W[(kk + 1) * D + col];
    packed[i] = b;
}

// One wave per 16-edge tile; 4 waves (tiles) per 128-thread block.
__global__ __launch_bounds__(128) void rgcn_wmma_kernel(
    const float* __restrict__ feat,    // [N, 64]
    const int*   __restrict__ src,     // [E]
    const int*   __restrict__ dst,     // [E]
    const int*   __restrict__ ws,      // counting-sort results
    const v2f*   __restrict__ wpk,     // packed B fragments
    float*       __restrict__ out)     // [N, 64]
{
    const int* rel_off  = ws + WS_RELOFF;
    const int* tile_off = ws + WS_TILEOFF;
    const int* sorted   = ws + WS_SORTED;

    int wave = threadIdx.x >> 5;
    int tile = blockIdx.x * 4 + wave;
    int totalTiles = tile_off[NREL];
    if (tile >= totalTiles) return;

    // Which relation does this tile belong to?
    int r = 0;
    while (r < NREL - 1 && tile >= tile_off[r + 1]) ++r;
    int tloc  = tile - tile_off[r];
    int ebase = rel_off[r] + tloc * 16;
    int ecnt  = rel_off[r + 1] - ebase;
    if (ecnt > 16) ecnt = 16;

    int  lane = threadIdx.x & 31;
    int  ml   = lane & 15;          // M (A) / N-in-tile (B,C) index
    bool hi   = lane >= 16;

    // --- A-fragment source row (per ISA 16x4 f32 A layout) ---
    bool avalid = ml < ecnt;
    int  ae     = avalid ? sorted[ebase + ml] : 0;
    long arowi  = avalid ? (long)src[ae] : 0;
    const float* arow   = feat + arowi * D;
    float        ascale = avalid ? 1.0f : 0.0f;   // zero rows of tail tiles

    // warm L2/WGP$ for this lane's feature row (256 B)
    __builtin_prefetch(arow, 0, 0);
    __builtin_prefetch(arow + 32, 0, 0);

    // Per-lane base into packed B fragments for relation r.
    const v2f* wp = wpk + (long)r * (16 * 4 * 32) + lane;

    v8f acc[4];
#pragma unroll
    for (int n = 0; n < 4; ++n)
        acc[n] = (v8f){0.f, 0.f, 0.f, 0.f, 0.f, 0.f, 0.f, 0.f};

    // 16 K-steps of K=4; 4 N-tiles of N=16 -> 64 v_wmma_f32_16x16x4_f32
#pragma unroll 4
    for (int k = 0; k < 16; ++k) {
        int kk = k * 4 + (hi ? 2 : 0);
        v2f a;
        a.x = arow[kk]     * ascale;
        a.y = arow[kk + 1] * ascale;
#pragma unroll
        for (int n = 0; n < 4; ++n) {
            v2f b = wp[k * 128 + n * 32];  // single coalesced b64 load
            acc[n] = __builtin_amdgcn_wmma_f32_16x16x4_f32(
                false, a, false, b, (short)0, acc[n], false, false);
        }
    }

    // --- Scatter-add C rows to out[dst] (C layout: VGPR j -> M=j / j+8) ---
#pragma unroll
    for (int j = 0; j < 8; ++j) {
        int m = j + (hi ? 8 : 0);
        if (m < ecnt) {
            int e = sorted[ebase + m];
            float* orow = out + (long)dst[e] * D;
#pragma unroll
            for (int n = 0; n < 4; ++n)
                atomicAdd(orow + n * 16 + ml, acc[n][j]);
        }
    }
}

extern "C" void kernel_launch(void* const* d_in, const int* in_sizes, int n_in,
                              void* d_out, int out_size, void* d_ws, size_t ws_size,
                              hipStream_t stream) {
    const float* feat   = (const float*)d_in[0];
    const float* weight = (const float*)d_in[1];
    const int*   src    = (const int*)d_in[2];
    const int*   dst    = (const int*)d_in[3];
    const int*   etypes = (const int*)d_in[4];
    float* out = (float*)d_out;
    int E = in_sizes[2];
    int* ws = (int*)d_ws;

    // packed weight fragments live after sorted_eid (8B aligned)
    size_t pk_off = ((size_t)(WS_SORTED + E) * sizeof(int) + 7) & ~(size_t)7;
    v2f* wpk = (v2f*)((char*)d_ws + pk_off);

    // 1. zero output accumulator + counting-sort state
    {
        int nblk = (out_size + 255) / 256;
        if (nblk > 2048) nblk = 2048;
        rgcn_zero_f32<<<nblk, 256, 0, stream>>>(out, out_size);
        rgcn_zero_i32<<<1, 64, 0, stream>>>(ws, 66);
    }
    // 2. pack weights into WMMA B-fragment order (runs concurrently-safe, same stream)
    rgcn_pack_w<<<(NREL * 16 * 4 * 32 + 255) / 256, 256, 0, stream>>>(weight, wpk);
    // 3. histogram over relations
    rgcn_hist<<<(E + 255) / 256, 256, 0, stream>>>(etypes, E, ws);
    // 4. exclusive scans (counts + tiles) and cursor reset
    rgcn_scan<<<1, 1, 0, stream>>>(ws);
    // 5. scatter edge ids into relation-sorted order
    rgcn_scatter<<<(E + 255) / 256, 256, 0, stream>>>(etypes, E, ws);
    // 6. segmented WMMA GEMM + scatter-add
    int maxTiles = (E + 15) / 16 + NREL;    // upper bound on sum(ceil(cnt_r/16))
    int blocks = (maxTiles + 3) / 4;        // 4 tiles (waves) per block
    rgcn_wmma_kernel<<<blocks, 128, 0, stream>>>(feat, src, dst, ws, wpk, out);
}